// MoELayer_37349035606098
// MI455X (gfx1250) — compile-verified
//
#include <hip/hip_runtime.h>
#include <cstddef>

typedef __attribute__((ext_vector_type(2))) float v2f;
typedef __attribute__((ext_vector_type(8))) float v8f;

#define D_DIM 1024
#define H_DIM 4096
#define E_NUM 8
#define HC    128     // H chunk per double-GEMM iteration
#define TM    16      // token tile (M)

__device__ __forceinline__ v8f wmma_f32_16x16x4(v2f a, v2f b, v8f c) {
    // 8 args: (neg_a, A, neg_b, B, c_mod, C, reuse_a, reuse_b)
    return __builtin_amdgcn_wmma_f32_16x16x4_f32(
        false, a, false, b, (short)0, c, false, false);
}

// ---------------------------------------------------------------------------
// Router: logits = x @ wg^T + bg ; top-2 ; softmax over the 2 ; append to
// per-expert buckets (token index + gate weight) with atomic counters.
// One wave per token; wg cached in LDS (32 KB).
// ---------------------------------------------------------------------------
__global__ __launch_bounds__(256) void moe_router(
    const float* __restrict__ x, const float* __restrict__ wg,
    const float* __restrict__ bg, int* __restrict__ counts,
    int* __restrict__ btok, float* __restrict__ bgate, int B)
{
    __shared__ float wgs[E_NUM * D_DIM];   // 32 KB
    for (int i = threadIdx.x; i < E_NUM * D_DIM; i += 256) wgs[i] = wg[i];
    __syncthreads();

    const int lane = threadIdx.x & 31;
    const int wv   = threadIdx.x >> 5;
    const int b    = blockIdx.x * 8 + wv;
    if (b >= B) return;

    const float* xr = x + (size_t)b * D_DIM;
    float acc[E_NUM];
#pragma unroll
    for (int e = 0; e < E_NUM; ++e) acc[e] = 0.0f;

    for (int d = lane; d < D_DIM; d += 32) {
        float xv = xr[d];
#pragma unroll
        for (int e = 0; e < E_NUM; ++e)
            acc[e] = fmaf(xv, wgs[e * D_DIM + d], acc[e]);
    }
#pragma unroll
    for (int e = 0; e < E_NUM; ++e) {
#pragma unroll
        for (int off = 16; off > 0; off >>= 1)
            acc[e] += __shfl_down(acc[e], off, 32);
    }

    if (lane == 0) {
        float lg[E_NUM];
#pragma unroll
        for (int e = 0; e < E_NUM; ++e) lg[e] = acc[e] + bg[e];
        int i0 = 0; float v0 = lg[0];
#pragma unroll
        for (int e = 1; e < E_NUM; ++e)
            if (lg[e] > v0) { v0 = lg[e]; i0 = e; }
        int i1 = (i0 == 0) ? 1 : 0; float v1 = lg[i1];
#pragma unroll
        for (int e = 0; e < E_NUM; ++e)
            if (e != i0 && lg[e] > v1) { v1 = lg[e]; i1 = e; }
        // softmax over the two selected logits (v0 >= v1)
        float e1 = __expf(v1 - v0);
        float g0 = 1.0f / (1.0f + e1);
        float g1 = e1 / (1.0f + e1);

        int p0 = atomicAdd(&counts[i0], 1);
        btok[(size_t)i0 * B + p0]  = b;
        bgate[(size_t)i0 * B + p0] = g0;
        int p1 = atomicAdd(&counts[i1], 1);
        btok[(size_t)i1 * B + p1]  = b;
        bgate[(size_t)i1 * B + p1] = g1;
    }
}

// ---------------------------------------------------------------------------
// Expert FFN: for a 16-token tile of one expert:
//   act = relu(X(16x1024) @ w1[e](1024xHC) + b1)    (stage 1, per HC chunk)
//   C  += act(16xHC) @ w2[e](HCx1024)               (stage 2, regs)
//   out[tok] += gate * (C + b2)                      (atomic scatter)
// 8 waves/block: stage1 -> each wave one 16-col tile of the HC=128 chunk;
// stage2 -> each wave 8 output tiles (128 of the 1024 output cols).
// ---------------------------------------------------------------------------
__global__ __launch_bounds__(256) void moe_ffn(
    const float* __restrict__ x,
    const float* __restrict__ w1, const float* __restrict__ b1,
    const float* __restrict__ w2, const float* __restrict__ b2,
    const int* __restrict__ counts, const int* __restrict__ btok,
    const float* __restrict__ bgate, float* __restrict__ out,
    int B, int mblocks)
{
    const int e      = blockIdx.x / mblocks;
    const int mb     = blockIdx.x % mblocks;
    const int cnt    = counts[e];
    const int m_base = mb * TM;
    if (m_base >= cnt) return;

    __shared__ float Xs[TM * D_DIM];   // 64 KB: gathered token rows
    __shared__ float As[TM * HC];      //  8 KB: relu activations chunk
    __shared__ int   tok_s[TM];
    __shared__ float gate_s[TM];

    const int tid = threadIdx.x;
    if (tid < TM) {
        int idx = m_base + tid;
        int tk = 0; float g = 0.0f;                 // padded rows: gate 0
        if (idx < cnt) {
            tk = btok[(size_t)e * B + idx];
            g  = bgate[(size_t)e * B + idx];
        }
        tok_s[tid] = tk; gate_s[tid] = g;
    }
    __syncthreads();

    // Gather 16 x-rows into LDS with 128-bit transfers.
    const float4* x4 = reinterpret_cast<const float4*>(x);
    for (int i = tid; i < TM * (D_DIM / 4); i += 256) {
        int row = i >> 8;            // D_DIM/4 == 256
        int c4  = i & 255;
        float4 v = x4[(size_t)tok_s[row] * (D_DIM / 4) + c4];
        *reinterpret_cast<float4*>(&Xs[row * D_DIM + c4 * 4]) = v;
    }
    __syncthreads();

    const int lane    = tid & 31;
    const int wv      = tid >> 5;
    const int col     = lane & 15;           // N (and A-row) index within tile
    const int koff    = (lane >> 4) << 1;    // 0 | 2 : k-pair per lane half
    const int halfsel = (lane >> 4) << 3;    // 0 | 8 : C-row offset per half

    const float* w1e = w1 + (size_t)e * D_DIM * H_DIM;
    const float* w2e = w2 + (size_t)e * H_DIM * D_DIM;
    const float* b1e = b1 + (size_t)e * H_DIM;
    const float* b2e = b2 + (size_t)e * D_DIM;

    const v8f vzero = {};
    v8f c[8];
#pragma unroll
    for (int t = 0; t < 8; ++t) c[t] = vzero;

    for (int hc0 = 0; hc0 < H_DIM; hc0 += HC) {
        // ---- stage 1: act tile (16 x 16) for columns [hc0+wv*16, +16) ----
        const int hcol = hc0 + wv * 16 + col;
        const float* w1p = w1e + hcol;
        v8f c1 = vzero;
#pragma unroll 4
        for (int kk = 0; kk < D_DIM; kk += 4) {
            v2f a = *reinterpret_cast<const v2f*>(&Xs[col * D_DIM + kk + koff]);
            v2f bb;
            size_t i0 = (size_t)(kk + koff) * H_DIM;
            bb.x = w1p[i0];
            bb.y = w1p[i0 + H_DIM];
            c1 = wmma_f32_16x16x4(a, bb, c1);
        }
        float bias1 = b1e[hcol];
#pragma unroll
        for (int r = 0; r < 8; ++r) c1[r] = fmaxf(c1[r] + bias1, 0.0f);

        __syncthreads();               // prev stage-2 done reading As
#pragma unroll
        for (int r = 0; r < 8; ++r)
            As[(r + halfsel) * HC + wv * 16 + col] = c1[r];
        __syncthreads();               // As chunk ready

        // ---- stage 2: C(16 x 128 per wave) += act @ w2 chunk ----
#pragma unroll 2
        for (int kk = 0; kk < HC; kk += 4) {
            v2f a = *reinterpret_cast<const v2f*>(&As[col * HC + kk + koff]);
            const float* w2p =
                w2e + (size_t)(hc0 + kk + koff) * D_DIM + wv * 128 + col;
#pragma unroll
            for (int t = 0; t < 8; ++t) {
                v2f bb;
                bb.x = w2p[t * 16];
                bb.y = w2p[t * 16 + D_DIM];
                c[t] = wmma_f32_16x16x4(a, bb, c[t]);
            }
        }
    }

    // ---- epilogue: out[tok] += gate * (C + b2) ----
#pragma unroll
    for (int t = 0; t < 8; ++t) {
        int n = wv * 128 + t * 16 + col;
        float bias2 = b2e[n];
#pragma unroll
        for (int r = 0; r < 8; ++r) {
            int m = r + halfsel;
            float val = gate_s[m] * (c[t][r] + bias2);
            atomicAdd(&out[(size_t)tok_s[m] * D_DIM + n], val);
        }
    }
}

// ---------------------------------------------------------------------------
extern "C" void kernel_launch(void* const* d_in, const int* in_sizes, int n_in,
                              void* d_out, int out_size, void* d_ws, size_t ws_size,
                              hipStream_t stream)
{
    const float* x  = (const float*)d_in[0];
    const float* wg = (const float*)d_in[1];
    const float* bg = (const float*)d_in[2];
    const float* w1 = (const float*)d_in[3];
    const float* b1 = (const float*)d_in[4];
    const float* w2 = (const float*)d_in[5];
    const float* b2 = (const float*)d_in[6];
    float* out = (float*)d_out;

    const int B = in_sizes[0] / D_DIM;

    // Workspace layout: [counts: 256B pad][btok: E*B int][bgate: E*B float]
    char*  ws     = (char*)d_ws;
    int*   counts = (int*)ws;
    int*   btok   = (int*)(ws + 256);
    float* bgate  = (float*)(ws + 256 + (size_t)E_NUM * B * sizeof(int));

    hipMemsetAsync(counts, 0, 256, stream);
    hipMemsetAsync(out, 0, (size_t)B * D_DIM * sizeof(float), stream);

    const int rblocks = (B + 7) / 8;
    moe_router<<<rblocks, 256, 0, stream>>>(x, wg, bg, counts, btok, bgate, B);

    const int mblocks = (B + TM - 1) / TM;
    moe_ffn<<<E_NUM * mblocks, 256, 0, stream>>>(
        x, w1, b1, w2, b2, counts, btok, bgate, out, B, mblocks);
}